// HiMoE_Adapter_163208757786
// MI455X (gfx1250) — compile-verified
//
#include <hip/hip_runtime.h>
#include <hip/hip_bf16.h>
#include <math.h>

typedef float v2f __attribute__((ext_vector_type(2)));
typedef float v8f __attribute__((ext_vector_type(8)));

#define C_DIM   768
#define E_NUM   7
#define R_RANK  8
#define KPAD    64          // E*R = 56 padded to 64
#define G1_KCHUNK 192       // K-chunk staged in LDS for GEMM1
#define EPS_F   2.2204460492503131e-16f

// -------------------------------------------------------------------------
// Kernel 1: gating.  One wave per token: logits = x[t] . w_gate (7 dots of
// length 768), wave-reduce, argmax (first-max to match jax top_k tie-break),
// write sel[t] and bump per-expert count.
// -------------------------------------------------------------------------
__global__ void __launch_bounds__(256)
gating_kernel(const float* __restrict__ x, const float* __restrict__ wg,
              int* __restrict__ sel, int* __restrict__ counts, int T) {
  int wave = threadIdx.x >> 5;
  int lane = threadIdx.x & 31;
  int t = blockIdx.x * (blockDim.x >> 5) + wave;
  if (t >= T) return;
  const float* xr = x + (size_t)t * C_DIM;
  float s[E_NUM];
#pragma unroll
  for (int e = 0; e < E_NUM; ++e) s[e] = 0.f;
  for (int k = lane; k < C_DIM; k += 32) {
    float xv = xr[k];
    const float* w = wg + k * E_NUM;
#pragma unroll
    for (int e = 0; e < E_NUM; ++e) s[e] += xv * w[e];
  }
#pragma unroll
  for (int off = 16; off > 0; off >>= 1) {
#pragma unroll
    for (int e = 0; e < E_NUM; ++e) s[e] += __shfl_xor(s[e], off);
  }
  if (lane == 0) {
    float best = s[0]; int bi = 0;
#pragma unroll
    for (int e = 1; e < E_NUM; ++e) {
      if (s[e] > best) { best = s[e]; bi = e; }   // strict > keeps first max
    }
    sel[t] = bi;
    atomicAdd(&counts[bi], 1);
  }
}

// -------------------------------------------------------------------------
// Kernel 2: loss = cv2(importance) + cv2(load); both equal the counts since
// K=1 gates are exactly 1.0 -> loss = 2 * cv2(counts).
// -------------------------------------------------------------------------
__global__ void loss_kernel(const int* __restrict__ counts,
                            float* __restrict__ out_loss) {
  float c[E_NUM], sum = 0.f;
#pragma unroll
  for (int e = 0; e < E_NUM; ++e) { c[e] = (float)counts[e]; sum += c[e]; }
  float mean = sum / (float)E_NUM;
  float ss = 0.f;
#pragma unroll
  for (int e = 0; e < E_NUM; ++e) { float d = c[e] - mean; ss += d * d; }
  float var = ss / (float)(E_NUM - 1);               // ddof=1
  float cv2 = var / (mean * mean + 1e-10f);
  *out_loss = 2.f * cv2;
}

// -------------------------------------------------------------------------
// Kernel 3: H[a][t][n] = sum_k x[t][k] * Wa[a][n/8][n%8][k],  n in [0,64)
// (n >= 56 zero-padded).  WMMA f32 16x16x4.  256 threads = 8 waves, each
// wave owns one 16-token M-tile; the shared B (Wa^T) K-chunk is staged in
// LDS in a pair-interleaved layout so each WMMA B operand is one ds_load_b64.
// -------------------------------------------------------------------------
__global__ void __launch_bounds__(256)
lora_a_gemm(const float* __restrict__ x, const float* __restrict__ Wa,
            float* __restrict__ H, int T) {
  // blds[(k>>1)*128 + n*2 + (k&1)] = WaT[k][n]
  __shared__ float blds[G1_KCHUNK * KPAD];           // 48 KB

  int a = blockIdx.y;
  const float* Wa_a = Wa + (size_t)a * E_NUM * R_RANK * C_DIM;
  int tid  = threadIdx.x;
  int lane = tid & 31;
  int wave = tid >> 5;
  int m = lane & 15;          // M row / N col within tile
  int h = lane >> 4;          // lane half -> K sub-pair
  int t0 = blockIdx.x * 128 + wave * 16;
  const float* xrow = x + (size_t)(t0 + m) * C_DIM;

  v8f acc0 = {}, acc1 = {}, acc2 = {}, acc3 = {};

  int fn  = tid & 63;                       // n this thread fills
  int fk0 = (tid >> 6) * (G1_KCHUNK / 4);   // 48 k-values per thread

  for (int chunk = 0; chunk < C_DIM / G1_KCHUNK; ++chunk) {
    // ---- stage WaT K-chunk into LDS (zeros for n >= 56) ----
    for (int i = 0; i < G1_KCHUNK / 4; i += 4) {
      int krel  = fk0 + i;
      int kglob = chunk * G1_KCHUNK + krel;
      float4 v;
      if (fn < E_NUM * R_RANK)
        v = *(const float4*)(Wa_a + (size_t)fn * C_DIM + kglob);
      else
        v = make_float4(0.f, 0.f, 0.f, 0.f);
      int base = (krel >> 1) * 128 + fn * 2;
      *(float2*)&blds[base]       = make_float2(v.x, v.y);
      *(float2*)&blds[base + 128] = make_float2(v.z, v.w);
    }
    __syncthreads();

    // ---- 48 WMMA K-steps over this chunk ----
    for (int s = 0; s < G1_KCHUNK / 4; ++s) {
      int kloc = 4 * s + 2 * h;
      v2f av = *(const v2f*)(xrow + chunk * G1_KCHUNK + kloc);
      const float* bb = &blds[kloc * KPAD];          // == (kloc>>1)*128
      v2f b0 = *(const v2f*)(bb + 2 * m);
      v2f b1 = *(const v2f*)(bb + 2 * (m + 16));
      v2f b2 = *(const v2f*)(bb + 2 * (m + 32));
      v2f b3 = *(const v2f*)(bb + 2 * (m + 48));
      acc0 = __builtin_amdgcn_wmma_f32_16x16x4_f32(false, av, false, b0, (short)0, acc0, false, false);
      acc1 = __builtin_amdgcn_wmma_f32_16x16x4_f32(false, av, false, b1, (short)0, acc1, false, false);
      acc2 = __builtin_amdgcn_wmma_f32_16x16x4_f32(false, av, false, b2, (short)0, acc2, false, false);
      acc3 = __builtin_amdgcn_wmma_f32_16x16x4_f32(false, av, false, b3, (short)0, acc3, false, false);
    }
    __syncthreads();
  }

  // ---- store H tile: D VGPR v holds row (v + 8*h), col m (+16j) ----
  float* Hbase = H + ((size_t)a * T + t0) * KPAD;
#pragma unroll
  for (int v = 0; v < 8; ++v) {
    float* hp = Hbase + (size_t)(v + 8 * h) * KPAD;
    hp[m]      = acc0[v];
    hp[m + 16] = acc1[v];
    hp[m + 32] = acc2[v];
    hp[m + 48] = acc3[v];
  }
}

// -------------------------------------------------------------------------
// Kernel 4: out[a][t][c] = log(max(exp( sum_k coeff(sel[t],k>>3)*H[a][t][k]
//                              * WbT[a][k][c] ), EPS))
// K=64 (8 pad cols are zero), N=768.  192 threads = 6 waves; each wave owns
// a 128-column slice of one 16-token M-tile.  Coefficient folded into the
// A operand (both k of a lane pair share the same expert since k is even).
// -------------------------------------------------------------------------
__global__ void __launch_bounds__(192)
lora_b_gemm(const float* __restrict__ H, const float* __restrict__ Wb,
            const int* __restrict__ sel, float* __restrict__ out, int T) {
  int a = blockIdx.y;
  const float* Wb_a = Wb + (size_t)a * E_NUM * C_DIM * R_RANK;
  int tid  = threadIdx.x;
  int lane = tid & 31;
  int wave = tid >> 5;       // 0..5 -> column slice
  int m = lane & 15;
  int h = lane >> 4;
  int t0 = blockIdx.x * 16;
  int wbase = wave * 128;

  const float* Hrow = H + ((size_t)a * T + t0 + m) * KPAD;
  int srow = sel[t0 + m];

  v8f acc[8];
#pragma unroll
  for (int j = 0; j < 8; ++j) acc[j] = (v8f){};

  for (int s = 0; s < KPAD / 4; ++s) {
    int k = 4 * s + 2 * h;
    int e = k >> 3;                      // expert of both k and k+1 (k even)
    v2f av = *(const v2f*)(Hrow + k);
    float cf = (e == srow) ? 1.f : 0.f;
    cf += (srow >= 4 && e < 4) ? 0.25f : 0.f;
    cf += (srow == 6 && (e == 4 || e == 5)) ? 0.5f : 0.f;
    av = av * cf;
    int ee = (e < E_NUM) ? e : (E_NUM - 1);   // pad region: coeff==0, clamp ptr
    int r = k & 7;                            // even
    const float* wb_e = Wb_a + (size_t)ee * C_DIM * R_RANK + r;
#pragma unroll
    for (int j = 0; j < 8; ++j) {
      int c = wbase + 16 * j + m;
      v2f bv = *(const v2f*)(wb_e + (size_t)c * R_RANK);
      acc[j] = __builtin_amdgcn_wmma_f32_16x16x4_f32(false, av, false, bv, (short)0, acc[j], false, false);
    }
  }

  // epilogue: y = log(exp(d)) with exp-underflow -> EPS (matches reference)
#pragma unroll
  for (int j = 0; j < 8; ++j) {
#pragma unroll
    for (int v = 0; v < 8; ++v) {
      float d  = acc[j][v];
      float ev = expf(d);
      ev = (ev == 0.f) ? EPS_F : ev;
      float y = logf(ev);
      int row = v + 8 * h;
      out[((size_t)a * T + t0 + row) * C_DIM + (wbase + 16 * j + m)] = y;
    }
  }
}

// -------------------------------------------------------------------------
extern "C" void kernel_launch(void* const* d_in, const int* in_sizes, int n_in,
                              void* d_out, int out_size, void* d_ws, size_t ws_size,
                              hipStream_t stream) {
  const float* x   = (const float*)d_in[0];   // [T, 768]
  const float* wg  = (const float*)d_in[1];   // [768, 7]
  const float* Wa  = (const float*)d_in[2];   // [3, 7, 8, 768]
  const float* Wb  = (const float*)d_in[3];   // [3, 7, 768, 8]
  float* out = (float*)d_out;                 // [3*T*768] ++ [1]

  int T = in_sizes[0] / C_DIM;                // 16384

  // workspace layout
  char* ws = (char*)d_ws;
  int*   sel    = (int*)ws;                                        // T ints
  size_t off    = ((size_t)T * sizeof(int) + 255) & ~(size_t)255;
  int*   counts = (int*)(ws + off);                                // 8 ints
  off += 256;
  float* H      = (float*)(ws + off);                              // [3][T][64]

  hipMemsetAsync(counts, 0, 8 * sizeof(int), stream);

  gating_kernel<<<dim3((T + 7) / 8), 256, 0, stream>>>(x, wg, sel, counts, T);
  loss_kernel<<<1, 1, 0, stream>>>(counts, out + (size_t)3 * T * C_DIM);

  dim3 g1(T / 128, 3);
  lora_a_gemm<<<g1, 256, 0, stream>>>(x, Wa, H, T);

  dim3 g2(T / 16, 3);
  lora_b_gemm<<<g2, 192, 0, stream>>>(H, Wb, sel, out, T);
}